// TempModel_62500364091414
// MI455X (gfx1250) — compile-verified
//
#include <hip/hip_runtime.h>
#include <stdint.h>

// Ensemble vote kernel for MI455X (gfx1250).
// logits: [8, B, 40] f32.  out: [B, 40] f32.
// - global_load_async_to_lds_b128 double-buffered staging (ASYNCcnt), per-wave
//   independent (no barriers): each wave stages only the 32 rows its own lanes consume.
// - argmax from LDS (argmax(softmax)==argmax(logits)), nibble-packed vote histogram,
//   softmax fallback from model 7's row.
// - output staged to LDS and drained with global_store_async_from_lds_b128
//   (coalesced 512B-per-instruction writes).

#define NMODELS   8
#define NCLASSES  40
#define ROWS_PER_WAVE 32
#define WAVES_PER_BLOCK 4
#define TILE_ROWS (ROWS_PER_WAVE * WAVES_PER_BLOCK)   // 128 rows / block
#define LDS_ROW_F 44                    // padded LDS row stride: 176 B (16B aligned)
#define WAVE_BUF_F (ROWS_PER_WAVE * LDS_ROW_F)        // 1408 floats per wave buffer
#define BUF_F (WAVES_PER_BLOCK * WAVE_BUF_F)          // 5632 floats per buffer half
#define CHUNKS_PER_WAVE 10              // 32 rows * 160 B / (32 lanes * 16 B)

__global__ __launch_bounds__(TILE_ROWS)
void ensemble_vote_kernel(const float* __restrict__ logits,
                          float* __restrict__ out,
                          int B)
{
    __shared__ float smem[2 * BUF_F];   // 45,056 B

    const int tid  = threadIdx.x;
    const int lane = tid & 31;
    const int w    = tid >> 5;
    const int rowBase = blockIdx.x * TILE_ROWS + w * ROWS_PER_WAVE; // wave's first row
    const size_t modelStride = (size_t)B * NCLASSES;                // floats

    // LDS byte addresses (low 32 bits of a flat LDS pointer == wave-relative offset)
    const uint32_t ldsBuf0 = (uint32_t)(uintptr_t)(&smem[0 * BUF_F + w * WAVE_BUF_F]);
    const uint32_t ldsBuf1 = (uint32_t)(uintptr_t)(&smem[1 * BUF_F + w * WAVE_BUF_F]);

    // Issue 10 async b128 loads covering this wave's 32x160B sub-tile for model m.
    auto issue = [&](int m, int buf) {
        const char* g = (const char*)(logits + (size_t)m * modelStride
                                             + (size_t)rowBase * NCLASSES);
        const uint32_t lb = buf ? ldsBuf1 : ldsBuf0;
#pragma unroll
        for (int i = 0; i < CHUNKS_PER_WAVE; ++i) {
            int k = i * 32 + lane;            // 16B chunk index, 0..319
            int q = (k * 6554) >> 16;         // k / 10 (row within sub-tile)
            int r = k - q * 10;               // chunk within row
            uint32_t ld = lb + (uint32_t)(q * 176 + r * 16);
            uint64_t ga = (uint64_t)(uintptr_t)(g + (size_t)k * 16);
            asm volatile("global_load_async_to_lds_b128 %0, %1, off"
                         :: "v"(ld), "v"(ga) : "memory");
        }
    };

    issue(0, 0);                              // prologue: prefetch model 0

    int   am[NMODELS];
    float rmax7 = 0.0f, rsum7 = 1.0f;

#pragma unroll
    for (int m = 0; m < NMODELS; ++m) {
        if (m + 1 < NMODELS) {
            // LDS reads of the buffer we are about to overwrite must be complete.
            asm volatile("s_wait_dscnt 0x0" ::: "memory");
            issue(m + 1, (m + 1) & 1);
            // 10 newest may be in flight; <=10 outstanding => model m's loads landed.
            asm volatile("s_wait_asynccnt 0xa" ::: "memory");
        } else {
            asm volatile("s_wait_asynccnt 0x0" ::: "memory");
        }

        const float* row = &smem[(size_t)(m & 1) * BUF_F + w * WAVE_BUF_F
                                 + lane * LDS_ROW_F];
        float best = row[0];
        int   bi   = 0;
#pragma unroll
        for (int c = 1; c < NCLASSES; ++c) {
            float v = row[c];
            if (v > best) { best = v; bi = c; }   // strict > keeps first max (jnp.argmax)
        }
        am[m] = bi;

        if (m == NMODELS - 1) {                   // model 7: softmax stats for fallback
            rmax7 = best;
            float s = 0.0f;
#pragma unroll
            for (int c = 0; c < NCLASSES; ++c) s += __expf(row[c] - rmax7);
            rsum7 = s;
        }
    }

    // Vote histogram, computed once: counts packed as 4-bit nibbles (max count = 8
    // would need 4 bits; counts 0..8, and 8 fits since nibble range is 0..15).
    uint32_t vp[5] = {0u, 0u, 0u, 0u, 0u};
    int maxv = 0;
#pragma unroll
    for (int c = 0; c < NCLASSES; ++c) {
        int vc = 0;
#pragma unroll
        for (int mm = 0; mm < NMODELS; ++mm) vc += (am[mm] == c) ? 1 : 0;
        maxv = vc > maxv ? vc : maxv;
        vp[c >> 3] |= (uint32_t)vc << ((c & 7) * 4);   // static indices after unroll
    }
    const bool  nom  = (maxv == 1);               // no majority -> softmax fallback
    const float invs = 1.0f / rsum7;
    const float* row7 = &smem[BUF_F + w * WAVE_BUF_F + lane * LDS_ROW_F]; // m=7: buf 1

    // Stage the output row into the (now free) buffer 0 region, then drain with
    // coalesced async stores.
    float* lrow = &smem[0 * BUF_F + w * WAVE_BUF_F + lane * LDS_ROW_F];
#pragma unroll
    for (int c0 = 0; c0 < NCLASSES; c0 += 4) {
        float4 o;
        float* op = (float*)&o;
#pragma unroll
        for (int j = 0; j < 4; ++j) {
            int c = c0 + j;
            int vc = (int)((vp[c >> 3] >> ((c & 7) * 4)) & 0xFu);
            float sm = __expf(row7[c] - rmax7) * invs;
            op[j] = nom ? sm : (float)vc;
        }
        *(float4*)(lrow + c0) = o;                // 16B-aligned LDS store
    }

    // LDS writes must land before the async store engine reads them.
    asm volatile("s_wait_dscnt 0x0" ::: "memory");

    const char* gout = (const char*)(out + (size_t)rowBase * NCLASSES);
#pragma unroll
    for (int i = 0; i < CHUNKS_PER_WAVE; ++i) {
        int k = i * 32 + lane;
        int q = (k * 6554) >> 16;
        int r = k - q * 10;
        uint32_t ls = ldsBuf0 + (uint32_t)(q * 176 + r * 16);
        uint64_t ga = (uint64_t)(uintptr_t)(gout + (size_t)k * 16);
        asm volatile("global_store_async_from_lds_b128 %0, %1, off"
                     :: "v"(ga), "v"(ls) : "memory");
    }
    // Drain before LDS dealloc / wave end (S_ENDPGM also does an implicit wait-idle).
    asm volatile("s_wait_asynccnt 0x0" ::: "memory");
}

// Tail path for rows not covered by full 128-row tiles (B=131072 divides evenly,
// but keep the kernel total-correct for any B). One thread per row, direct loads.
__global__ void ensemble_vote_tail_kernel(const float* __restrict__ logits,
                                          float* __restrict__ out,
                                          int B, int rowStart)
{
    int row = rowStart + blockIdx.x * blockDim.x + threadIdx.x;
    if (row >= B) return;
    const size_t modelStride = (size_t)B * NCLASSES;

    int am[NMODELS];
    float rmax7 = 0.0f, rsum7 = 1.0f;
#pragma unroll
    for (int m = 0; m < NMODELS; ++m) {
        const float* r = logits + (size_t)m * modelStride + (size_t)row * NCLASSES;
        float best = r[0]; int bi = 0;
#pragma unroll
        for (int c = 1; c < NCLASSES; ++c) {
            float v = r[c];
            if (v > best) { best = v; bi = c; }
        }
        am[m] = bi;
        if (m == NMODELS - 1) {
            rmax7 = best;
            float s = 0.0f;
#pragma unroll
            for (int c = 0; c < NCLASSES; ++c) s += __expf(r[c] - rmax7);
            rsum7 = s;
        }
    }
    uint32_t vp[5] = {0u, 0u, 0u, 0u, 0u};
    int maxv = 0;
#pragma unroll
    for (int c = 0; c < NCLASSES; ++c) {
        int vc = 0;
#pragma unroll
        for (int mm = 0; mm < NMODELS; ++mm) vc += (am[mm] == c) ? 1 : 0;
        maxv = vc > maxv ? vc : maxv;
        vp[c >> 3] |= (uint32_t)vc << ((c & 7) * 4);
    }
    const bool  nom  = (maxv == 1);
    const float invs = 1.0f / rsum7;
    const float* r7  = logits + (size_t)(NMODELS - 1) * modelStride + (size_t)row * NCLASSES;
    float* orow = out + (size_t)row * NCLASSES;
#pragma unroll
    for (int c = 0; c < NCLASSES; ++c) {
        int vc = (int)((vp[c >> 3] >> ((c & 7) * 4)) & 0xFu);
        float sm = __expf(r7[c] - rmax7) * invs;
        orow[c] = nom ? sm : (float)vc;
    }
}

extern "C" void kernel_launch(void* const* d_in, const int* in_sizes, int n_in,
                              void* d_out, int out_size, void* d_ws, size_t ws_size,
                              hipStream_t stream)
{
    const float* logits = (const float*)d_in[0];
    float* out = (float*)d_out;

    const int total = in_sizes[0];                 // 8 * B * 40
    const int B     = total / (NMODELS * NCLASSES);

    const int tiles = B / TILE_ROWS;               // full 128-row tiles
    if (tiles > 0) {
        ensemble_vote_kernel<<<tiles, TILE_ROWS, 0, stream>>>(logits, out, B);
    }
    const int rem = B - tiles * TILE_ROWS;
    if (rem > 0) {
        const int rowStart = tiles * TILE_ROWS;
        const int blk = 128;
        ensemble_vote_tail_kernel<<<(rem + blk - 1) / blk, blk, 0, stream>>>(
            logits, out, B, rowStart);
    }
}